// ModelPytorch_82068235092129
// MI455X (gfx1250) — compile-verified
//
#include <hip/hip_runtime.h>
#include <cmath>

typedef __attribute__((ext_vector_type(2))) float v2f;
typedef __attribute__((ext_vector_type(8))) float v8f;

#define RNN_B 1024
#define RNN_T 512
#define RNN_H 128
#define RNN_P 4
#define RNN_TT (RNN_T + RNN_P)   // 516 output timesteps
#define NTILE 8                  // N tiles of 16 (H = 128)
#define KCHUNK 32                // K chunks of 4 covering H = 128
#define HB_STRIDE 132            // hbuf row stride (floats): bank-conflict-free

// fcw pack: B fragment for (kc, nt) = 2 floats per lane at
//   fcw[((kc*8 + nt)*32 + lane)*2 + {0,1}]
// where value(j) = fc_w[n*H + k], n = nt*16 + lane%16, k = 4*kc + 2*(lane/16) + j.
#define FCW_ELEMS (KCHUNK * NTILE * 32 * 2)   // 16384 floats = 64 KB

__device__ __forceinline__ float dev_tanh(float v) {
#if __has_builtin(__builtin_amdgcn_tanhf)
  return __builtin_amdgcn_tanhf(v);
#else
  return tanhf(v);
#endif
}

__global__ __launch_bounds__(32) void rnn516_wmma(
    const float* __restrict__ x, const float* __restrict__ hidden,
    const float* __restrict__ fc0_w, const float* __restrict__ fc0_b,
    const float* __restrict__ fc_w, const float* __restrict__ fc_b,
    const float* __restrict__ dec_w, const float* __restrict__ dec_b,
    float* __restrict__ y) {
  extern __shared__ float lds[];
  float* fcw  = lds;                          // fragment-packed fc_w: 16384 f
  float* hbuf = lds + FCW_ELEMS;              // [16][HB_STRIDE]
  float* ybuf = hbuf + 16 * HB_STRIDE;        // [16]

  const int lane = threadIdx.x;               // 0..31 (wave32)
  const int half = lane >> 4;                 // 0 | 1
  const int lmod = lane & 15;
  const int b0 = blockIdx.x * 16;             // this wave's 16 batch rows

  // ---- stage fc_w into LDS, fragment-packed for conflict-free ds_load_b64
  for (int i = lane; i < FCW_ELEMS; i += 32) {
    int j  = i & 1;
    int l2 = (i >> 1) & 31;
    int nt = (i >> 6) & 7;
    int kc = i >> 9;
    int n = nt * 16 + (l2 & 15);
    int k = 4 * kc + 2 * (l2 >> 4) + j;
    fcw[i] = fc_w[n * RNN_H + k];
  }
  // ---- stage initial hidden state (row-major, padded stride)
  for (int i = lane; i < 16 * HB_STRIDE; i += 32) {
    int r = i / HB_STRIDE, k = i - r * HB_STRIDE;
    hbuf[i] = (k < RNN_H) ? hidden[(b0 + r) * RNN_H + k] : 0.0f;
  }
  __syncthreads();

  // per-lane, per-N-tile constants
  float biasv[NTILE], dwv[NTILE];
  v2f blast[NTILE];  // B fragment for the x * fc0_w rank-1 update
#pragma unroll
  for (int nt = 0; nt < NTILE; ++nt) {
    int n = nt * 16 + lmod;
    biasv[nt] = fc0_b[n] + fc_b[n];
    dwv[nt] = dec_w[n];
    v2f b;
    b[0] = (half == 0) ? fc0_w[n] : 0.0f;  // B[k=0][n] = fc0_w[n]
    b[1] = 0.0f;
    blast[nt] = b;
  }
  const float decb = dec_b[0];

  float xval = 0.0f;  // lanes 0-15: x_t for row (b0+lmod); lanes 16-31 stay 0

  for (int t = 0; t < RNN_TT; ++t) {
    if (t < RNN_T) {
      if (half == 0) xval = x[(b0 + lmod) * RNN_T + t];
    }  // else: xval = previous decoded y (autoregressive)

    // accumulators initialized with combined bias (C = bias[n] everywhere)
    v8f d[NTILE];
#pragma unroll
    for (int nt = 0; nt < NTILE; ++nt) {
#pragma unroll
      for (int e = 0; e < 8; ++e) d[nt][e] = biasv[nt];
    }

    // ---- main K loop, software-pipelined by one kc so LDS latency
    //      overlaps the matrix pipe instead of s_wait_dscnt 0 stalls
    const float* arow  = hbuf + lmod * HB_STRIDE + 2 * half;
    const float* bcol  = fcw + 2 * lane;

    v2f a_cur, b_cur[NTILE], a_nxt, b_nxt[NTILE];
    a_cur = *(const v2f*)(arow);
#pragma unroll
    for (int nt = 0; nt < NTILE; ++nt)
      b_cur[nt] = *(const v2f*)(bcol + nt * 64);

#pragma unroll 4
    for (int kc = 0; kc < KCHUNK - 1; ++kc) {
      // prefetch kc+1 fragments (issued before consuming kc's)
      a_nxt = *(const v2f*)(arow + 4 * (kc + 1));
      const float* bb = bcol + (kc + 1) * (NTILE * 64);
#pragma unroll
      for (int nt = 0; nt < NTILE; ++nt)
        b_nxt[nt] = *(const v2f*)(bb + nt * 64);

#pragma unroll
      for (int nt = 0; nt < NTILE; ++nt)
        d[nt] = __builtin_amdgcn_wmma_f32_16x16x4_f32(
            false, a_cur, false, b_cur[nt], (short)0, d[nt], false, false);

      a_cur = a_nxt;
#pragma unroll
      for (int nt = 0; nt < NTILE; ++nt) b_cur[nt] = b_nxt[nt];
    }
    // epilogue: last K chunk
#pragma unroll
    for (int nt = 0; nt < NTILE; ++nt)
      d[nt] = __builtin_amdgcn_wmma_f32_16x16x4_f32(
          false, a_cur, false, b_cur[nt], (short)0, d[nt], false, false);

    // rank-1 input term: A = [x_m, 0, 0, 0], B = [fc0_w[n]; 0; 0; 0]
    {
      v2f al;
      al[0] = xval;  // lanes 16-31 hold 0 -> A[m][2] = 0
      al[1] = 0.0f;
#pragma unroll
      for (int nt = 0; nt < NTILE; ++nt)
        d[nt] = __builtin_amdgcn_wmma_f32_16x16x4_f32(
            false, al, false, blast[nt], (short)0, d[nt], false, false);
    }

    // tanh, write back h (transposed store into hbuf), accumulate decode
    float yred[8];
#pragma unroll
    for (int v = 0; v < 8; ++v) yred[v] = 0.0f;
#pragma unroll
    for (int nt = 0; nt < NTILE; ++nt) {
      int col = nt * 16 + lmod;
#pragma unroll
      for (int v = 0; v < 8; ++v) {
        float th = dev_tanh(d[nt][v]);
        hbuf[(v + 8 * half) * HB_STRIDE + col] = th;  // row m = v + 8*half
        yred[v] += th * dwv[nt];
      }
    }

    // reduce y over the 16 lanes of each half (rows v / v+8)
#pragma unroll
    for (int v = 0; v < 8; ++v) {
      float s = yred[v];
      s += __shfl_xor(s, 1, 32);
      s += __shfl_xor(s, 2, 32);
      s += __shfl_xor(s, 4, 32);
      s += __shfl_xor(s, 8, 32);
      if (lmod == 0) ybuf[v + 8 * half] = s + decb;
    }

    float yv = ybuf[lmod];  // same-wave LDS ops are in-order: no barrier
    if (half == 0) {
      y[(b0 + lmod) * RNN_TT + t] = yv;
      xval = yv;  // feeds the 4 autoregressive steps
    }
  }
}

extern "C" void kernel_launch(void* const* d_in, const int* in_sizes, int n_in,
                              void* d_out, int out_size, void* d_ws,
                              size_t ws_size, hipStream_t stream) {
  const float* x      = (const float*)d_in[0];
  const float* hidden = (const float*)d_in[1];
  const float* fc0_w  = (const float*)d_in[2];
  const float* fc0_b  = (const float*)d_in[3];
  const float* fc_w   = (const float*)d_in[4];
  const float* fc_b   = (const float*)d_in[5];
  const float* dec_w  = (const float*)d_in[6];
  const float* dec_b  = (const float*)d_in[7];
  float* y = (float*)d_out;

  const size_t shmem =
      (size_t)(FCW_ELEMS + 16 * HB_STRIDE + 16) * sizeof(float);  // ~74 KB
  rnn516_wmma<<<dim3(RNN_B / 16), dim3(32), shmem, stream>>>(
      x, hidden, fc0_w, fc0_b, fc_w, fc_b, dec_w, dec_b, y);
}